// SuperpointGenerator_28372553957928
// MI455X (gfx1250) — compile-verified
//
#include <hip/hip_runtime.h>

#define NPTS  4096
#define WPR   128          // 4096/32 adjacency words per row
#define NB    8            // batches
#define SENT  4096
#define EPS2  0.0036f      // 0.06^2
#define MAXSP 512

typedef __attribute__((ext_vector_type(2))) float v2f;
typedef __attribute__((ext_vector_type(8))) float v8f;
typedef __attribute__((ext_vector_type(4))) unsigned tdm_g0_t;   // D# group 0
typedef __attribute__((ext_vector_type(8))) int      tdm_g1_t;   // D# group 1
typedef __attribute__((ext_vector_type(4))) int      tdm_g2_t;   // D# groups 2/3
typedef __attribute__((ext_vector_type(8))) int      tdm_g4_t;   // trailing group (6-arg form)

// ---------------------------------------------------------------------------
// K0: pack each point into a 16B record pk = (x, y, z, ||p||^2).
// ---------------------------------------------------------------------------
__global__ __launch_bounds__(256) void pack_kernel(const float* __restrict__ coords,
                                                   float4* __restrict__ pk) {
  const int gid = blockIdx.x * 256 + threadIdx.x;      // 0..NB*NPTS-1
  const float x = coords[gid * 3 + 0];
  const float y = coords[gid * 3 + 1];
  const float z = coords[gid * 3 + 2];
  pk[gid] = make_float4(x, y, z, x * x + y * y + z * z);
}

// ---------------------------------------------------------------------------
// K1: adjacency bitmask via WMMA f32 16x16x4.
// Workgroup = 4 waves = 4 row-tiles of one batch. The batch's packed points
// (64 KB) are staged into LDS once via the Tensor Data Mover, then each wave
// loops over 128 strips of 32 columns feeding WMMA from LDS.
// D = A x B + C with A row m = (x,y,z,1), B col n = (-2x,-2y,-2z,||q||^2),
// C[m][*] = ||p_m||^2  =>  D[m][n] = d^2(p_m, q_n). Ballot compresses the
// 16x16 f32 tile to adjacency bits in-register.
// ---------------------------------------------------------------------------
__global__ __launch_bounds__(128) void adj_kernel(const float4* __restrict__ pk,
                                                  unsigned* __restrict__ adj) {
  __shared__ float4 spk[NPTS];                         // 64 KB, LDS offset 0

  const int wid  = blockIdx.x * 4 + (threadIdx.x >> 5);   // 0..2047
  const int lane = threadIdx.x & 31;
  const int b    = blockIdx.x >> 6;                       // 64 blocks per batch
  const int ti   = wid & 255;                             // 16-row tile
  unsigned* ab   = adj + (size_t)b * NPTS * WPR;

  // ---- TDM: stage pk[b] (4096 x 16B = 8192 x 8B) into LDS, wave 0 only ----
  if (threadIdx.x < 32) {
    const unsigned long long ga =
        (unsigned long long)(uintptr_t)(pk + (size_t)b * NPTS);
    const unsigned nelem = NPTS * 2u;                  // 8192 8-byte elements

    tdm_g0_t g0;
    g0[0] = 1u;                                        // count=1, user mode
    g0[1] = 0u;                                        // lds_addr = 0 (spk)
    g0[2] = (unsigned)(ga & 0xFFFFFFFFu);              // global_addr[31:0]
    g0[3] = (unsigned)((ga >> 32) & 0x1FFFFFFu) | (2u << 30);  // addr[56:32], type=2

    tdm_g1_t g1;
    g1[0] = (int)(3u << 16);                           // wg_mask=0, data_size=8B
    g1[1] = (int)((nelem & 0xFFFFu) << 16);            // tensor_dim0[15:0]
    g1[2] = (int)(((nelem >> 16) & 0xFFFFu) | (1u << 16)); // dim0[31:16], tensor_dim1=1
    g1[3] = (int)((nelem & 0xFFFFu) << 16);            // dim1 hi=0, tile_dim0=8192
    g1[4] = 1;                                         // tile_dim1=1, tile_dim2=0
    g1[5] = (int)nelem;                                // tensor_dim0_stride[31:0]
    g1[6] = 0;                                         // stride hi, dim1_stride lo
    g1[7] = 0;

    tdm_g2_t gz = {0, 0, 0, 0};
    tdm_g4_t gz8 = {0, 0, 0, 0, 0, 0, 0, 0};
    __builtin_amdgcn_tensor_load_to_lds(g0, g1, gz, gz, gz8, 0);
    __builtin_amdgcn_s_wait_tensorcnt(0);
  }
  __syncthreads();
  // The TDM write to LDS is invisible to the optimizer (the target is an
  // integer offset inside the descriptor). Force it to treat spk as written
  // so the loads below stay real ds_load_b128 instead of folding to undef.
  asm volatile("" : : "r"(&spk[0]) : "memory");

  const int h = lane >> 4;        // half-wave (selects K pair)
  const int m = lane & 15;        // row within tile / col within tile

  const float4 rp4 = spk[ti * 16 + m];
  const float  sqr = rp4.w;

  // A 16x4 layout: lanes 0-15 hold K=0,1 in v0,v1; lanes 16-31 hold K=2,3.
  v2f a;
  a.x = h ? rp4.z : rp4.x;
  a.y = h ? 1.0f  : rp4.y;

  // C/D layout: VGPR r -> rows (r, r+8) across half-waves. Preload ||p||^2.
  v8f c;
#pragma unroll
  for (int r = 0; r < 8; ++r) c[r] = __shfl(sqr, r + 8 * h, 32);

  for (int tj = 0; tj < WPR; ++tj) {
    const float4 q0 = spk[tj * 32 + m];
    const float4 q1 = spk[tj * 32 + 16 + m];

    v2f b0, b1;                       // B 4x16: lanes 0-15 K=0,1; lanes 16-31 K=2,3
    b0.x = h ? -2.0f * q0.z : -2.0f * q0.x;
    b0.y = h ? q0.w         : -2.0f * q0.y;
    b1.x = h ? -2.0f * q1.z : -2.0f * q1.x;
    b1.y = h ? q1.w         : -2.0f * q1.y;

    v8f d0 = __builtin_amdgcn_wmma_f32_16x16x4_f32(false, a, false, b0,
                                                   (short)0, c, false, false);
    v8f d1 = __builtin_amdgcn_wmma_f32_16x16x4_f32(false, a, false, b1,
                                                   (short)0, c, false, false);

    unsigned lo[8], hi[8];
#pragma unroll
    for (int r = 0; r < 8; ++r) {
      unsigned t0 = (unsigned)__ballot(d0[r] <= EPS2);
      unsigned t1 = (unsigned)__ballot(d1[r] <= EPS2);
      lo[r] = (t0 & 0xFFFFu) | (t1 << 16);          // row r   : cols tj*32..+31
      hi[r] = (t0 >> 16) | (t1 & 0xFFFF0000u);      // row r+8 : cols tj*32..+31
    }
    if (lane == 0) {
      unsigned* rp = ab + (size_t)(ti * 16) * WPR + tj;
#pragma unroll
      for (int r = 0; r < 8; ++r) rp[(size_t)r * WPR] = lo[r];
#pragma unroll
      for (int r = 0; r < 8; ++r) rp[(size_t)(r + 8) * WPR] = hi[r];
    }
  }
}

// ---------------------------------------------------------------------------
// K2: degree -> core flags, packed as a bitmask (ballot per 32 points).
// Diagonal is set in adj (d2(i,i)=0), matching the reference's degree count.
// ---------------------------------------------------------------------------
__global__ __launch_bounds__(256) void core_kernel(const unsigned* __restrict__ adj,
                                                   unsigned* __restrict__ coremask) {
  const int gid = blockIdx.x * 256 + threadIdx.x;   // 0..NB*NPTS-1
  const unsigned* rp = adj + (size_t)gid * WPR;
  int cnt = 0;
  for (int w = 0; w < WPR; ++w) cnt += __popc(rp[w]);
  const unsigned msk = (unsigned)__ballot(cnt >= 3);
  if ((threadIdx.x & 31) == 0) coremask[gid >> 5] = msk;
}

// ---------------------------------------------------------------------------
// K3: one workgroup per batch. Label propagation + pointer jumping in LDS,
// then border/noise fill, compaction, and top-512 cap, exactly mirroring
// the reference's semantics. Labels written as float32.
// ---------------------------------------------------------------------------
__global__ __launch_bounds__(1024) void label_kernel(const unsigned* __restrict__ adj,
                                                     const unsigned* __restrict__ coremask,
                                                     float* __restrict__ out) {
  __shared__ int l[NPTS];        // current labels
  __shared__ int ln[NPTS];       // next labels / remap tables / present+rank
  __shared__ int scr[NPTS];      // counts / scan partials / counts2
  __shared__ unsigned cm[WPR];   // core bitmask
  __shared__ int s_flag, s_any, s_mk;

  const int t = threadIdx.x;
  const int b = blockIdx.x;
  const unsigned* ab = adj + (size_t)b * NPTS * WPR;

  if (t < WPR) cm[t] = coremask[b * WPR + t];
  __syncthreads();

  // l0 = core ? idx : SENT
  for (int k = 0; k < 4; ++k) {
    int i = t + k * 1024;
    l[i] = ((cm[i >> 5] >> (i & 31)) & 1) ? i : SENT;
  }
  __syncthreads();

  // ---- min-label propagation with pointer jumping ----
  for (;;) {
    if (t == 0) s_flag = 0;
    __syncthreads();
    int oldl[4];
    for (int k = 0; k < 4; ++k) {
      int i = t + k * 1024;
      int li = l[i];
      oldl[k] = li;
      int v;
      if ((cm[i >> 5] >> (i & 31)) & 1) {
        int nb = li;                                  // includes self (adj diag set)
        const unsigned* rp = ab + (size_t)i * WPR;
        for (int w = 0; w < WPR; ++w) {
          unsigned mw = rp[w] & cm[w];                // core neighbors only
          while (mw) {
            int j = (w << 5) + __builtin_ctz(mw);
            mw &= mw - 1;
            int lj = l[j];
            nb = lj < nb ? lj : nb;
          }
        }
        v = nb;
      } else {
        v = SENT;
      }
      ln[i] = v;
    }
    __syncthreads();
    int chg = 0;
    for (int k = 0; k < 4; ++k) {                     // pointer jump on snapshot ln
      int i = t + k * 1024;
      int v = ln[i];
      if (v < SENT) { int lv = ln[v]; v = lv < v ? lv : v; }
      l[i] = v;
      chg |= (v != oldl[k]);
    }
    if (chg) atomicOr(&s_flag, 1);
    __syncthreads();
    int f = s_flag;
    __syncthreads();
    if (!f) break;
  }

  // ---- border points: min core-neighbor label; core keep their label ----
  for (int k = 0; k < 4; ++k) {
    int i = t + k * 1024;
    int v;
    if ((cm[i >> 5] >> (i & 31)) & 1) {
      v = l[i];
    } else {
      int nb = SENT;
      const unsigned* rp = ab + (size_t)i * WPR;
      for (int w = 0; w < WPR; ++w) {
        unsigned mw = rp[w] & cm[w];
        while (mw) {
          int j = (w << 5) + __builtin_ctz(mw);
          mw &= mw - 1;
          int lj = l[j];
          nb = lj < nb ? lj : nb;
        }
      }
      v = nb;
    }
    ln[i] = v;
  }
  __syncthreads();
  for (int k = 0; k < 4; ++k) { int i = t + k * 1024; l[i] = ln[i]; }
  for (int k = 0; k < 4; ++k) { int i = t + k * 1024; scr[i] = 0; }
  if (t == 0) { s_any = 0; s_mk = -1; }
  __syncthreads();

  // ---- counts of non-noise labels; most-common via packed-key argmax ----
  int anyloc = 0;
  for (int k = 0; k < 4; ++k) {
    int i = t + k * 1024;
    int li = l[i];
    if (li < SENT) { atomicAdd(&scr[li], 1); anyloc = 1; }
  }
  if (anyloc) atomicOr(&s_any, 1);
  __syncthreads();
  int best = -1;
  for (int k = 0; k < 4; ++k) {
    int i = t + k * 1024;
    int key = (scr[i] << 12) | (4095 - i);            // max count, first index wins
    best = key > best ? key : best;
  }
  atomicMax(&s_mk, best);
  __syncthreads();
  const int mc   = 4095 - (s_mk & 4095);
  const int anyv = s_any;
  for (int k = 0; k < 4; ++k) {
    int i = t + k * 1024;
    if (l[i] == SENT) l[i] = anyv ? mc : 0;
  }
  __syncthreads();

  // ---- compaction: present -> rank (inclusive scan - 1) ----
  for (int k = 0; k < 4; ++k) { int i = t + k * 1024; ln[i] = 0; }
  __syncthreads();
  for (int k = 0; k < 4; ++k) { int i = t + k * 1024; ln[l[i]] = 1; }
  __syncthreads();
  const int p0 = ln[4 * t], p1 = ln[4 * t + 1], p2 = ln[4 * t + 2], p3 = ln[4 * t + 3];
  const int s1 = p0 + p1, s2 = s1 + p2, s3 = s2 + p3;
  scr[t] = s3;
  __syncthreads();
  for (int off = 1; off < 1024; off <<= 1) {          // Hillis-Steele inclusive scan
    int v = (t >= off) ? scr[t - off] : 0;
    __syncthreads();
    scr[t] += v;
    __syncthreads();
  }
  const int base = t ? scr[t - 1] : 0;
  const int nclu = scr[1023];
  ln[4 * t + 0] = base + p0 - 1;
  ln[4 * t + 1] = base + s1 - 1;
  ln[4 * t + 2] = base + s2 - 1;
  ln[4 * t + 3] = base + s3 - 1;
  __syncthreads();
  for (int k = 0; k < 4; ++k) { int i = t + k * 1024; l[i] = ln[l[i]]; }
  __syncthreads();

  // ---- counts2 and top-512 cap (stable-argsort semantics via key ranking) ----
  for (int k = 0; k < 4; ++k) { int i = t + k * 1024; scr[i] = 0; }
  __syncthreads();
  for (int k = 0; k < 4; ++k) { int i = t + k * 1024; atomicAdd(&scr[l[i]], 1); }
  __syncthreads();
  if (nclu > MAXSP) {                                  // uniform branch per block
    for (int k = 0; k < 4; ++k) {
      int i = t + k * 1024;
      int keyi = (scr[i] << 12) | i;
      int rd = 0;
      for (int j = 0; j < NPTS; ++j) {
        int kj = (scr[j] << 12) | j;
        rd += (kj > keyi);
      }
      ln[i] = (rd < MAXSP) ? (MAXSP - 1 - rd) : 0;     // remap[top] = arange(512)
    }
    __syncthreads();
    for (int k = 0; k < 4; ++k) { int i = t + k * 1024; l[i] = ln[l[i]]; }
    __syncthreads();
  }

  for (int k = 0; k < 4; ++k) {
    int i = t + k * 1024;
    out[(size_t)b * NPTS + i] = (float)l[i];
  }
}

// ---------------------------------------------------------------------------
extern "C" void kernel_launch(void* const* d_in, const int* in_sizes, int n_in,
                              void* d_out, int out_size, void* d_ws, size_t ws_size,
                              hipStream_t stream) {
  (void)in_sizes; (void)n_in; (void)out_size; (void)ws_size;
  const float* coords = (const float*)d_in[0];        // [8,4096,3] f32
  float* out          = (float*)d_out;                // [8,4096] labels as f32

  // workspace: adjacency bitmask (16 MB) + core bitmask (4 KB) + packed pts
  unsigned* adj      = (unsigned*)d_ws;
  unsigned* coremask = adj + (size_t)NB * NPTS * WPR;
  float4*   pk       = (float4*)(coremask + NB * WPR);

  pack_kernel <<<dim3(NB * NPTS / 256), dim3(256),  0, stream>>>(coords, pk);
  adj_kernel  <<<dim3(NB * 256 / 4),    dim3(128),  0, stream>>>(pk, adj);
  core_kernel <<<dim3(NB * NPTS / 256), dim3(256),  0, stream>>>(adj, coremask);
  label_kernel<<<dim3(NB),              dim3(1024), 0, stream>>>(adj, coremask, out);
}